// VNStdFeature_66726611911208
// MI455X (gfx1250) — compile-verified
//
#include <hip/hip_runtime.h>

typedef __attribute__((ext_vector_type(2))) float v2f;
typedef __attribute__((ext_vector_type(8))) float v8f;

#define NWAVES 8      // waves per block
#define TP     16     // points per wave (WMMA N tile)

// LDS row strides padded so stride % 64 == 2 -> conflict-free b64 row fetches
#define W1S 258
#define D1S 130
#define W2S 130
#define D2S 66
#define W3S 66
#define BUF128 130    // col-major activation buffer stride (128-row stages)
#define BUF64  66     // col-major activation buffer stride (64-row stages)

#define OFF_W1  0
#define OFF_D1  (OFF_W1 + 128*W1S)
#define OFF_W2  (OFF_D1 + 128*D1S)
#define OFF_D2  (OFF_W2 + 64*W2S)
#define OFF_W3  (OFF_D2 + 64*D2S)
#define OFF_BUF (OFF_W3 + 16*W3S)
#define BUF_FLOATS (16*BUF128)
#define OFF_Z   (OFF_BUF + NWAVES*BUF_FLOATS)
#define LDS_FLOATS (OFF_Z + NWAVES*144)   // 81056 floats = 324224 B <= 320KB WGP LDS

static __device__ inline v8f wmma4(v2f a, v2f b, v8f c) {
  // D = A(16x4,f32) * B(4x16,f32) + C(16x16,f32)
  return __builtin_amdgcn_wmma_f32_16x16x4_f32(false, a, false, b, (short)0, c,
                                               false, false);
}

// GEMM over K with A (weights, row-major, stride WS) and B (activations,
// col-major, stride BS) both in LDS.  MT = number of 16-row M tiles.
template<int MT, int K>
static __device__ inline void gemm_lds(const float* __restrict__ W, int WS,
                                       const float* __restrict__ B, int BS,
                                       v8f* acc, int l16, int half) {
#pragma unroll 2
  for (int k = 0; k < K; k += 4) {
    const int c0 = k + 2*half;                 // VGPR0 holds k_rel=2h, VGPR1 k_rel=2h+1
    v2f b = *(const v2f*)(B + l16*BS + c0);
#pragma unroll
    for (int mt = 0; mt < MT; ++mt) {
      v2f a = *(const v2f*)(W + (mt*16 + l16)*WS + c0);
      acc[mt] = wmma4(a, b, acc[mt]);
    }
  }
}

// Spill a D-layout accumulator group to a col-major LDS buffer (stride S).
// Lane (l16, half): VGPR r holds row r + 8*half -> pairs (2i,2i+1) are b64s.
template<int MT>
static __device__ inline void spill_cm(const v8f* acc, float* buf, int S,
                                       int l16, int half) {
#pragma unroll
  for (int mt = 0; mt < MT; ++mt) {
#pragma unroll
    for (int i = 0; i < 4; ++i) {
      v2f p;
      p.x = acc[mt][2*i];
      p.y = acc[mt][2*i + 1];
      *(v2f*)(buf + l16*S + mt*16 + 8*half + 2*i) = p;
    }
  }
}

// VN leaky-relu applied in WMMA D-layout: column reduction over M via
// per-lane partials + xor-16 shuffle (lanes L / L+16 hold complementary rows).
template<int MT>
static __device__ inline void vn_nonlin(v8f* y, const v8f* d) {
  float dot = 0.f, dns = 0.f;
#pragma unroll
  for (int mt = 0; mt < MT; ++mt) {
#pragma unroll
    for (int i = 0; i < 8; ++i) {
      dot += y[mt][i] * d[mt][i];
      dns += d[mt][i] * d[mt][i];
    }
  }
  dot += __shfl_xor(dot, 16, 32);
  dns += __shfl_xor(dns, 16, 32);
  // mask=1 -> z=y ; mask=0 -> z = y - 0.8*(dot/(|d|^2+eps))*d
  const float s = (dot >= 0.f) ? 0.f : 0.8f * dot / (dns + 1e-7f);
#pragma unroll
  for (int mt = 0; mt < MT; ++mt) {
#pragma unroll
    for (int i = 0; i < 8; ++i) {
      y[mt][i] = y[mt][i] - s * d[mt][i];
    }
  }
}

__global__ void __launch_bounds__(256, 1)
vnstd_kernel(const float* __restrict__ xg,
             const float* __restrict__ W1g, const float* __restrict__ D1g,
             const float* __restrict__ W2g, const float* __restrict__ D2g,
             const float* __restrict__ W3g,
             float* __restrict__ xstd, float* __restrict__ z0out) {
  extern __shared__ float smem[];
  const int tid = threadIdx.x;

  // ---- cooperative weight staging (padded rows) ----
  for (int i = tid; i < 128*256; i += 256) smem[OFF_W1 + (i >> 8)*W1S + (i & 255)] = W1g[i];
  for (int i = tid; i < 128*128; i += 256) smem[OFF_D1 + (i >> 7)*D1S + (i & 127)] = D1g[i];
  for (int i = tid; i < 64*128;  i += 256) smem[OFF_W2 + (i >> 7)*W2S + (i & 127)] = W2g[i];
  for (int i = tid; i < 64*64;   i += 256) smem[OFF_D2 + (i >> 6)*D2S + (i & 63)]  = D2g[i];
  for (int i = tid; i < 16*64;   i += 256) // W3 zero-padded to 16 rows
    smem[OFF_W3 + (i >> 6)*W3S + (i & 63)] = (i < 3*64) ? W3g[i] : 0.f;
  __syncthreads();

  const int wave = tid >> 5;
  const int lane = tid & 31;
  const int l16  = lane & 15;
  const int half = lane >> 4;

  float* buf = smem + OFF_BUF + wave * BUF_FLOATS;   // wave-private activation scratch
  float* zb  = smem + OFF_Z   + wave * 144;          // wave-private z (3t x 3r x 16p)

  const int p0 = (blockIdx.x * NWAVES + wave) * TP;  // base point of this wave's tile
  const int gp = p0 + l16;                           // this lane's B-operand point

  const float* sW1 = smem + OFF_W1;
  const float* sD1 = smem + OFF_D1;
  const float* sW2 = smem + OFF_W2;
  const float* sD2 = smem + OFF_D2;
  const float* sW3 = smem + OFF_W3;

  const v8f vzero = {0.f, 0.f, 0.f, 0.f, 0.f, 0.f, 0.f, 0.f};

#pragma unroll 1
  for (int t = 0; t < 3; ++t) {
    // ---- stage 1: y1 = W1 @ x_t  (128 x 16), B streamed from global ----
    v8f y1[8];
#pragma unroll
    for (int mt = 0; mt < 8; ++mt) y1[mt] = vzero;

    const float* xb = xg + (size_t)gp * 768 + t;     // channel stride 3 floats
#pragma unroll 2
    for (int k = 0; k < 256; k += 4) {
      const int c0 = k + 2*half;
      v2f b;
      b.x = xb[c0*3];
      b.y = xb[c0*3 + 3];
#pragma unroll
      for (int mt = 0; mt < 8; ++mt) {
        v2f a = *(const v2f*)(sW1 + (mt*16 + l16)*W1S + c0);
        y1[mt] = wmma4(a, b, y1[mt]);
      }
    }
    spill_cm<8>(y1, buf, BUF128, l16, half);

    // ---- stage 2: d1 = D1 @ y1 ----
    v8f d1[8];
#pragma unroll
    for (int mt = 0; mt < 8; ++mt) d1[mt] = vzero;
    gemm_lds<8, 128>(sD1, D1S, buf, BUF128, d1, l16, half);

    vn_nonlin<8>(y1, d1);                  // z1 now in y1
    spill_cm<8>(y1, buf, BUF128, l16, half);

    // ---- stage 3: y2 = W2 @ z1  (64 x 16) ----
    v8f y2[4];
#pragma unroll
    for (int mt = 0; mt < 4; ++mt) y2[mt] = vzero;
    gemm_lds<4, 128>(sW2, W2S, buf, BUF128, y2, l16, half);
    spill_cm<4>(y2, buf, BUF64, l16, half);

    // ---- stage 4: d2 = D2 @ y2 ----
    v8f d2[4];
#pragma unroll
    for (int mt = 0; mt < 4; ++mt) d2[mt] = vzero;
    gemm_lds<4, 64>(sD2, D2S, buf, BUF64, d2, l16, half);

    vn_nonlin<4>(y2, d2);                  // z2 now in y2
    spill_cm<4>(y2, buf, BUF64, l16, half);

    // ---- stage 5: z_t = W3pad @ z2  (rows 0..2 valid) ----
    v8f zt = vzero;
    gemm_lds<1, 64>(sW3, W3S, buf, BUF64, &zt, l16, half);

    if (half == 0) {                       // rows 0..2 live in VGPR 0..2, lanes 0-15
#pragma unroll
      for (int r = 0; r < 3; ++r) zb[(t*3 + r)*16 + l16] = zt[r];
    }
  }

  // ---- final: x_std = x_p @ z0, z0 = z^T ; z[k][j] = zb[(j*3+k)*16 + p] ----
#pragma unroll 1
  for (int p = 0; p < TP; ++p) {
    const int pg = p0 + p;
    float zm[3][3];                        // zm[j][k] = z0[j][k]
#pragma unroll
    for (int j = 0; j < 3; ++j)
#pragma unroll
      for (int k = 0; k < 3; ++k) zm[j][k] = zb[(j*3 + k)*16 + p];

    const float* xp = xg + (size_t)pg * 768;   // L2-hot re-read
    float* op = xstd + (size_t)pg * 768;
#pragma unroll 1
    for (int cb = 0; cb < 8; ++cb) {
      const int c = cb*32 + lane;
      const float x0 = xp[c*3 + 0];
      const float x1 = xp[c*3 + 1];
      const float x2 = xp[c*3 + 2];
#pragma unroll
      for (int k = 0; k < 3; ++k)
        op[c*3 + k] = x0*zm[0][k] + x1*zm[1][k] + x2*zm[2][k];
    }
  }

  // ---- z0 output: [B,N,3,3], z0[p][j][k] = zb[(j*3+k)*16 + p] ----
  if (lane < 16) {
    const int pg = p0 + lane;
#pragma unroll
    for (int j = 0; j < 3; ++j)
#pragma unroll
      for (int k = 0; k < 3; ++k)
        z0out[(size_t)pg*9 + j*3 + k] = zb[(j*3 + k)*16 + lane];
  }
}

extern "C" void kernel_launch(void* const* d_in, const int* in_sizes, int n_in,
                              void* d_out, int out_size, void* d_ws, size_t ws_size,
                              hipStream_t stream) {
  (void)in_sizes; (void)n_in; (void)out_size; (void)d_ws; (void)ws_size;
  const float* x  = (const float*)d_in[0];
  const float* W1 = (const float*)d_in[1];
  const float* D1 = (const float*)d_in[2];
  const float* W2 = (const float*)d_in[3];
  const float* D2 = (const float*)d_in[4];
  const float* W3 = (const float*)d_in[5];

  float* xstd = (float*)d_out;
  float* z0   = xstd + (size_t)8 * 4096 * 256 * 3;   // 25165824 floats

  const int points = 8 * 4096;                       // 32768
  const int blocks = points / (NWAVES * TP);         // 256
  const size_t shmem = (size_t)LDS_FLOATS * sizeof(float);  // 324224 B

  vnstd_kernel<<<dim3(blocks), dim3(256), shmem, stream>>>(
      x, W1, D1, W2, D2, W3, xstd, z0);
}